// CRF_43379169689964
// MI455X (gfx1250) — compile-verified
//
#include <hip/hip_runtime.h>
#include <cstddef>
#include <cstdint>

// CRF Viterbi decode: B=32 batches, T=1024 steps, K=512 tags.
// Forward: tropical (max,+) recurrence, states written into d_out.
// Backward: recompute backpointers row-by-row, overwrite d_out with one-hot.

namespace {
constexpr int kB = 32;
constexpr int kT = 1024;
constexpr int kK = 512;
}  // namespace

// ---- CDNA5 async-to-LDS support (guarded; falls back to sync copy) --------
#if defined(__gfx1250__) && __has_builtin(__builtin_amdgcn_global_load_async_to_lds_b32) && \
    __has_builtin(__builtin_amdgcn_s_wait_asynccnt)
#define CRF_ASYNC 1
#else
#define CRF_ASYNC 0
#endif

__device__ __forceinline__ void crf_async_ld(const float* g, float* l) {
#if CRF_ASYNC
  typedef __attribute__((address_space(1))) int GI;  // prints as "__device__ int *"
  typedef __attribute__((address_space(3))) int LI;  // prints as "__shared__ int *"
  __builtin_amdgcn_global_load_async_to_lds_b32((GI*)(int*)g, (LI*)(int*)l, 0, 0);
#else
  *l = *g;
#endif
}

#if CRF_ASYNC
#define CRF_WAIT_ASYNC(n) __builtin_amdgcn_s_wait_asynccnt(n)
#else
#define CRF_WAIT_ASYNC(n) ((void)0)
#endif

// ---------------------------------------------------------------------------
// Forward pass: one block per batch, 512 threads = 128 j-groups x 4 i-chunks.
// Each thread covers 4 consecutive j (one float4 trans load per i) over a
// 128-wide i-chunk; partial maxes reduced through LDS. States (max-only, no
// argmax) are written to states[b][t][k] (== d_out, reused as scratch).
// ---------------------------------------------------------------------------
__global__ __launch_bounds__(512) void crf_fwd(const float* __restrict__ pot,
                                               const float* __restrict__ trans,
                                               float* __restrict__ states) {
  const int b = blockIdx.x;
  const int tid = (int)threadIdx.x;
  const int ic = tid >> 7;    // i-chunk 0..3 (128 rows each)
  const int jg = tid & 127;   // j-group
  const int jb = jg << 2;     // first of 4 output tags handled
  const int ibase = ic << 7;

  __shared__ float s_state[kK];          // 2 KB: state[t-1]
  __shared__ float s_red[4][128][4];     // 8 KB: partial maxes [ic][jg][comp]
  __shared__ float s_pot[2][kK];         // 4 KB: double-buffered pot row

  const float* potb = pot + (size_t)b * kT * kK;
  float* stb = states + (size_t)b * kT * kK;

  // t = 0: state = pot[0]
  const float st0 = potb[tid];
  s_state[tid] = st0;
  stb[tid] = st0;
  // Prefetch pot row 1 into buffer 1 (async on CDNA5).
  crf_async_ld(&potb[(size_t)kK + tid], &s_pot[1][tid]);
  __syncthreads();

  for (int t = 1; t < kT; ++t) {
    // Kick off the async prefetch of pot row t+1 while we compute step t.
    if (t + 1 < kT)
      crf_async_ld(&potb[(size_t)(t + 1) * kK + tid], &s_pot[(t + 1) & 1][tid]);

    const float* trow = trans + (size_t)ibase * kK + jb;
    float m0 = -__FLT_MAX__, m1 = -__FLT_MAX__, m2 = -__FLT_MAX__, m3 = -__FLT_MAX__;
#pragma unroll 8
    for (int ii = 0; ii < 128; ++ii) {
      const float s = s_state[ibase + ii];  // uniform across the wave -> LDS broadcast
      const float4 tv = *reinterpret_cast<const float4*>(trow + (size_t)ii * kK);
      m0 = fmaxf(m0, s + tv.x);
      m1 = fmaxf(m1, s + tv.y);
      m2 = fmaxf(m2, s + tv.z);
      m3 = fmaxf(m3, s + tv.w);
    }
    *reinterpret_cast<float4*>(&s_red[ic][jg][0]) = make_float4(m0, m1, m2, m3);

    // Row t's async load (issued last iteration) must be done; row t+1 may
    // still be in flight (1 outstanding). Last iteration has nothing in
    // flight beyond row t, so drain fully.
    if (t + 1 < kT) {
      CRF_WAIT_ASYNC(1);
    } else {
      CRF_WAIT_ASYNC(0);
    }
    __syncthreads();

    // Combine the 4 i-chunk partials; thread tid owns tag j = tid.
    const int rjg = tid >> 2;
    const int comp = tid & 3;
    float v = fmaxf(fmaxf(s_red[0][rjg][comp], s_red[1][rjg][comp]),
                    fmaxf(s_red[2][rjg][comp], s_red[3][rjg][comp]));
    v += s_pot[t & 1][tid];

    s_state[tid] = v;                    // distinct array from s_red: no race
    stb[(size_t)t * kK + tid] = v;       // persist state row for the backward pass
    __syncthreads();
  }
}

// ---------------------------------------------------------------------------
// Block-wide argmax over 512 lanes, first-index tie-break (matches jnp.argmax).
// wave32 shuffle reduce, then 16 wave results combined in wave 0.
// ---------------------------------------------------------------------------
__device__ __forceinline__ void crf_amax2(float& v, int& i, float v2, int i2) {
  if (v2 > v || (v2 == v && i2 < i)) { v = v2; i = i2; }
}

__device__ __forceinline__ int crf_block_argmax(float v, int idx, float* s_v,
                                                int* s_i, int* s_res) {
  const int tid = (int)threadIdx.x;
  const int lane = tid & 31;
  const int wv = tid >> 5;
#pragma unroll
  for (int m = 16; m >= 1; m >>= 1) {
    const float v2 = __shfl_xor(v, m, 32);
    const int i2 = __shfl_xor(idx, m, 32);
    crf_amax2(v, idx, v2, i2);
  }
  if (lane == 0) { s_v[wv] = v; s_i[wv] = idx; }
  __syncthreads();
  if (wv == 0) {
    float vv = s_v[lane & 15];  // lanes 16..31 replicate: harmless for argmax
    int ii = s_i[lane & 15];
#pragma unroll
    for (int m = 8; m >= 1; m >>= 1) {
      const float v2 = __shfl_xor(vv, m, 32);
      const int i2 = __shfl_xor(ii, m, 32);
      crf_amax2(vv, ii, v2, i2);
    }
    if (lane == 0) *s_res = ii;
  }
  __syncthreads();
  const int r = *s_res;
  __syncthreads();
  return r;
}

// ---------------------------------------------------------------------------
// Backward pass: one block per batch. Recompute the backpointer for the one
// chosen tag per step (O(K) work), then overwrite each state row in `out`
// with its one-hot encoding. Row t-1 is always fully read before it is
// overwritten (the argmax barriers order it).
// ---------------------------------------------------------------------------
__global__ __launch_bounds__(512) void crf_bwd(const float* __restrict__ trans,
                                               const float* __restrict__ transT,
                                               int use_transT,
                                               float* __restrict__ out) {
  const int b = blockIdx.x;
  const int tid = (int)threadIdx.x;
  float* ob = out + (size_t)b * kT * kK;

  __shared__ float s_v[16];
  __shared__ int s_i[16];
  __shared__ int s_res;

  // last_tag = argmax over the final state row.
  const float vlast = ob[(size_t)(kT - 1) * kK + tid];
  int tag = crf_block_argmax(vlast, tid, s_v, s_i, &s_res);

  for (int t = kT - 1; t >= 1; --t) {
    const float tr = use_transT ? transT[(size_t)tag * kK + tid]   // coalesced
                                : trans[(size_t)tid * kK + tag];   // strided fallback
    const float sc = ob[(size_t)(t - 1) * kK + tid] + tr;
    const int prev = crf_block_argmax(sc, tid, s_v, s_i, &s_res);
    ob[(size_t)t * kK + tid] = (tid == tag) ? 1.0f : 0.0f;  // one-hot row t
    tag = prev;
  }
  ob[tid] = (tid == tag) ? 1.0f : 0.0f;  // one-hot row 0
}

// ---------------------------------------------------------------------------
// 512x512 transpose of the transition matrix into d_ws so the backward pass
// reads columns coalesced.
// ---------------------------------------------------------------------------
__global__ __launch_bounds__(256) void crf_transpose(const float* __restrict__ in,
                                                     float* __restrict__ out) {
  __shared__ float tile[32][33];
  const int bx = (int)blockIdx.x, by = (int)blockIdx.y;
  const int tx = (int)threadIdx.x, ty = (int)threadIdx.y;  // 32 x 8
  const int x = bx * 32 + tx;
#pragma unroll
  for (int k = 0; k < 32; k += 8) {
    const int y = by * 32 + ty + k;
    tile[ty + k][tx] = in[(size_t)y * kK + x];
  }
  __syncthreads();
  const int x2 = by * 32 + tx;
#pragma unroll
  for (int k = 0; k < 32; k += 8) {
    const int y2 = bx * 32 + ty + k;
    out[(size_t)y2 * kK + x2] = tile[tx][ty + k];
  }
}

// ---------------------------------------------------------------------------
extern "C" void kernel_launch(void* const* d_in, const int* in_sizes, int n_in,
                              void* d_out, int out_size, void* d_ws, size_t ws_size,
                              hipStream_t stream) {
  (void)in_sizes; (void)n_in; (void)out_size;
  const float* pot = (const float*)d_in[0];    // [B,T,K] f32
  const float* trans = (const float*)d_in[1];  // [K,K]  f32
  float* out = (float*)d_out;                  // [B,T,K] f32 one-hot

  const size_t transT_bytes = (size_t)kK * kK * sizeof(float);
  const bool useT = (d_ws != nullptr) && (ws_size >= transT_bytes);
  float* transT = useT ? (float*)d_ws : nullptr;

  if (useT) {
    dim3 g(kK / 32, kK / 32), blk(32, 8);
    crf_transpose<<<g, blk, 0, stream>>>(trans, transT);
  }
  crf_fwd<<<kB, 512, 0, stream>>>(pot, trans, out);
  crf_bwd<<<kB, 512, 0, stream>>>(trans, transT, useT ? 1 : 0, out);
}